// SetTransformerCritic_18202071401087
// MI455X (gfx1250) — compile-verified
//
#include <hip/hip_runtime.h>
#include <math.h>

// ---------------------------------------------------------------------------
// SetTransformerCritic forward for gfx1250 (MI455X).
// GEMMs use V_WMMA_F32_16X16X4_F32 (f32 in/out, exact-precision path),
// 32x64 tile per wave, register-pipelined fragment loads.
// Attention is flash-style per (b,h,q) row in LDS -> no HBM score matrices.
// ---------------------------------------------------------------------------

typedef __attribute__((ext_vector_type(2))) float v2f;
typedef __attribute__((ext_vector_type(8))) float v8f;

__device__ __forceinline__ float gelu_f(float x) {
  return 0.5f * x * (1.0f + erff(x * 0.70710678118654752f));
}

__device__ __forceinline__ float blk_sum(float v, float* red) {
  int tid = threadIdx.x;
  red[tid] = v; __syncthreads();
  for (int st = blockDim.x >> 1; st > 0; st >>= 1) {
    if (tid < st) red[tid] += red[tid + st];
    __syncthreads();
  }
  float r = red[0]; __syncthreads();
  return r;
}

__device__ __forceinline__ float blk_max(float v, float* red) {
  int tid = threadIdx.x;
  red[tid] = v; __syncthreads();
  for (int st = blockDim.x >> 1; st > 0; st >>= 1) {
    if (tid < st) red[tid] = fmaxf(red[tid], red[tid + st]);
    __syncthreads();
  }
  float r = red[0]; __syncthreads();
  return r;
}

#define WMMA_F32(a, b, c) \
  __builtin_amdgcn_wmma_f32_16x16x4_f32(false, (a), false, (b), (short)0, (c), false, false)

// ---------------------------------------------------------------------------
// WMMA GEMM: C[M,N] = act(A[M,K] @ W[N,K]^T + bias[N])
// One wave (32 threads) per 32x64 output tile; f32 WMMA 16x16x4.
// A-frag: lane L holds A[m0+(L&15)][kk + 2*(L>>4) + {0,1}]  (ISA 16x4 layout)
// B-frag: lane L holds W[n0+(L&15)][kk + 2*(L>>4) + {0,1}]  (B = W^T, 4x16)
// C/D:    VGPR r -> row m0 + r + 8*(L>>4), col n0 + (L&15)
// Fragment loads are software-pipelined one k-step ahead so WMMAs overlap
// the next iteration's global loads; __launch_bounds__(32,1) lifts the VGPR
// cap so a0/a1 and the next-iteration fragments keep distinct registers.
// K % 4 == 0, N % 64 == 0, M % 32 == 0 (true for every call below).
// ---------------------------------------------------------------------------
__global__ void __launch_bounds__(32, 1)
gemm_wmma_kernel(const float* __restrict__ A,
                 const float* __restrict__ W,
                 const float* __restrict__ bias,
                 float* __restrict__ C,
                 int M, int N, int K, int act) {
  int lane = threadIdx.x;            // 0..31, full wave, EXEC all ones
  int m0 = blockIdx.y << 5;          // 32 rows
  int n0 = blockIdx.x << 6;          // 64 cols
  int r = lane & 15;
  int koff = (lane >> 4) << 1;       // 0 or 2

  const float* ap0 = A + (size_t)(m0 + 0  + r) * K + koff;
  const float* ap1 = A + (size_t)(m0 + 16 + r) * K + koff;
  const float* wp0 = W + (size_t)(n0 + 0  + r) * K + koff;
  const float* wp1 = W + (size_t)(n0 + 16 + r) * K + koff;
  const float* wp2 = W + (size_t)(n0 + 32 + r) * K + koff;
  const float* wp3 = W + (size_t)(n0 + 48 + r) * K + koff;

  v8f z = {0.f,0.f,0.f,0.f,0.f,0.f,0.f,0.f};
  v8f acc00 = z, acc01 = z, acc02 = z, acc03 = z;
  v8f acc10 = z, acc11 = z, acc12 = z, acc13 = z;

  // prologue loads (k-step 0)
  v2f a0 = *(const v2f*)(ap0);
  v2f a1 = *(const v2f*)(ap1);
  v2f b0 = *(const v2f*)(wp0);
  v2f b1 = *(const v2f*)(wp1);
  v2f b2 = *(const v2f*)(wp2);
  v2f b3 = *(const v2f*)(wp3);

  for (int kk = 4; kk < K; kk += 4) {
    // issue next iteration's loads before consuming the current fragments
    v2f na0 = *(const v2f*)(ap0 + kk);
    v2f na1 = *(const v2f*)(ap1 + kk);
    v2f nb0 = *(const v2f*)(wp0 + kk);
    v2f nb1 = *(const v2f*)(wp1 + kk);
    v2f nb2 = *(const v2f*)(wp2 + kk);
    v2f nb3 = *(const v2f*)(wp3 + kk);

    acc00 = WMMA_F32(a0, b0, acc00);
    acc01 = WMMA_F32(a0, b1, acc01);
    acc02 = WMMA_F32(a0, b2, acc02);
    acc03 = WMMA_F32(a0, b3, acc03);
    acc10 = WMMA_F32(a1, b0, acc10);
    acc11 = WMMA_F32(a1, b1, acc11);
    acc12 = WMMA_F32(a1, b2, acc12);
    acc13 = WMMA_F32(a1, b3, acc13);

    a0 = na0; a1 = na1; b0 = nb0; b1 = nb1; b2 = nb2; b3 = nb3;
  }
  acc00 = WMMA_F32(a0, b0, acc00);
  acc01 = WMMA_F32(a0, b1, acc01);
  acc02 = WMMA_F32(a0, b2, acc02);
  acc03 = WMMA_F32(a0, b3, acc03);
  acc10 = WMMA_F32(a1, b0, acc10);
  acc11 = WMMA_F32(a1, b1, acc11);
  acc12 = WMMA_F32(a1, b2, acc12);
  acc13 = WMMA_F32(a1, b3, acc13);

  int mhi = (lane >> 4) << 3;
  int col = lane & 15;
  float bb0 = bias[n0 + 0  + col];
  float bb1 = bias[n0 + 16 + col];
  float bb2 = bias[n0 + 32 + col];
  float bb3 = bias[n0 + 48 + col];
  #pragma unroll
  for (int rr = 0; rr < 8; ++rr) {
    int ma = m0 + rr + mhi;
    int mb = m0 + 16 + rr + mhi;
    float u0 = acc00[rr] + bb0, u1 = acc01[rr] + bb1;
    float u2 = acc02[rr] + bb2, u3 = acc03[rr] + bb3;
    float w0 = acc10[rr] + bb0, w1 = acc11[rr] + bb1;
    float w2 = acc12[rr] + bb2, w3 = acc13[rr] + bb3;
    if (act) {
      u0 = gelu_f(u0); u1 = gelu_f(u1); u2 = gelu_f(u2); u3 = gelu_f(u3);
      w0 = gelu_f(w0); w1 = gelu_f(w1); w2 = gelu_f(w2); w3 = gelu_f(w3);
    }
    float* ca = C + (size_t)ma * N + n0 + col;
    float* cb = C + (size_t)mb * N + n0 + col;
    ca[0] = u0; ca[16] = u1; ca[32] = u2; ca[48] = u3;
    cb[0] = w0; cb[16] = w1; cb[32] = w2; cb[48] = w3;
  }
}

// ---------------------------------------------------------------------------
// Attention: one block (256 thr) per (b, h, q). Scores for all Nk keys in LDS,
// softmax, weighted sum of V. biasMode: 0=none, 1=2*(conf-0.5), 2=conf.
// Q:[B,Nq,256] K,V:[B,Nk,256], head h = cols [h*32, h*32+32). Nk <= 1024.
// ---------------------------------------------------------------------------
__global__ void attn_kernel(const float* __restrict__ Q, const float* __restrict__ K,
                            const float* __restrict__ V, const float* __restrict__ conf,
                            const int* __restrict__ mask, float* __restrict__ O,
                            int Nq, int Nk, int biasMode) {
  __shared__ float sarr[1024];
  __shared__ float red[256];
  __shared__ float qs[32];
  __shared__ float part[256];

  int bid = blockIdx.x;
  int qi = bid % Nq;
  int h  = (bid / Nq) & 7;
  int b  = bid / (Nq * 8);
  int tid = threadIdx.x;

  const float* qrow = Q + ((size_t)(b * Nq + qi)) * 256 + h * 32;
  if (tid < 32) qs[tid] = qrow[tid];
  __syncthreads();

  float lmax = -3.0e38f;
  for (int k0 = tid; k0 < Nk; k0 += 256) {
    const float4* krow = (const float4*)(K + ((size_t)(b * Nk + k0)) * 256 + h * 32);
    float s = 0.f;
    #pragma unroll
    for (int d4 = 0; d4 < 8; ++d4) {
      float4 kv = krow[d4];
      s += qs[4 * d4 + 0] * kv.x + qs[4 * d4 + 1] * kv.y +
           qs[4 * d4 + 2] * kv.z + qs[4 * d4 + 3] * kv.w;
    }
    s *= 0.17677669529663687f;                  // 1/sqrt(32)
    float c = conf[b * Nk + k0];
    if (biasMode == 1) s += 2.f * (c - 0.5f);
    else if (biasMode == 2) s += c;
    if (mask[b * Nk + k0] == 0) s = -1.0e9f;
    sarr[k0] = s;
    lmax = fmaxf(lmax, s);
  }
  float m = blk_max(lmax, red);

  float lsum = 0.f;
  for (int k0 = tid; k0 < Nk; k0 += 256) {
    float e = expf(sarr[k0] - m);
    sarr[k0] = e;
    lsum += e;
  }
  float ssum = blk_sum(lsum, red);
  float inv = 1.f / ssum;

  int d = tid & 31, chunk = tid >> 5;
  float acc = 0.f;
  for (int k0 = chunk; k0 < Nk; k0 += 8)
    acc += sarr[k0] * V[((size_t)(b * Nk + k0)) * 256 + h * 32 + d];
  part[tid] = acc;
  __syncthreads();
  if (tid < 32) {
    float o = 0.f;
    #pragma unroll
    for (int c2 = 0; c2 < 8; ++c2) o += part[c2 * 32 + tid];
    O[((size_t)(b * Nq + qi)) * 256 + h * 32 + tid] = o * inv;
  }
}

// LN over D=256 with optional residual: O = LN(X + R). 256 threads per row.
__global__ void ln_kernel(const float* __restrict__ X, const float* __restrict__ R,
                          const float* __restrict__ g, const float* __restrict__ bb,
                          float* __restrict__ O) {
  __shared__ float red[256];
  int row = blockIdx.x, tid = threadIdx.x;
  float v = X[(size_t)row * 256 + tid];
  if (R) v += R[(size_t)row * 256 + tid];
  float mean = blk_sum(v, red) * (1.f / 256.f);
  float dd = v - mean;
  float var = blk_sum(dd * dd, red) * (1.f / 256.f);
  O[(size_t)row * 256 + tid] = dd * rsqrtf(var + 1e-5f) * g[tid] + bb[tid];
}

// project(): LN(gelu(l2(LN(gelu(l1(x)))))). din<=7, 128 threads per row.
__global__ void proj_kernel(const float* __restrict__ X, int din,
                            const float* __restrict__ w1, const float* __restrict__ b1,
                            const float* __restrict__ g1, const float* __restrict__ bb1,
                            const float* __restrict__ w2, const float* __restrict__ b2,
                            const float* __restrict__ g2, const float* __restrict__ bb2,
                            float* __restrict__ O) {
  __shared__ float xs[8];
  __shared__ float hh[128];
  __shared__ float red[128];
  int row = blockIdx.x, tid = threadIdx.x;      // 128 threads
  if (tid < din) xs[tid] = X[(size_t)row * din + tid];
  __syncthreads();

  float s = b1[tid];
  for (int i = 0; i < din; ++i) s += w1[tid * din + i] * xs[i];
  s = gelu_f(s);
  float mean = blk_sum(s, red) * (1.f / 128.f);
  float dd = s - mean;
  float var = blk_sum(dd * dd, red) * (1.f / 128.f);
  hh[tid] = dd * rsqrtf(var + 1e-5f) * g1[tid] + bb1[tid];
  __syncthreads();

  float o0 = b2[tid], o1 = b2[tid + 128];
  for (int i = 0; i < 128; ++i) {
    float hv = hh[i];
    o0 += w2[(size_t)tid * 128 + i] * hv;
    o1 += w2[(size_t)(tid + 128) * 128 + i] * hv;
  }
  o0 = gelu_f(o0); o1 = gelu_f(o1);
  float mean2 = blk_sum(o0 + o1, red) * (1.f / 256.f);
  float d0 = o0 - mean2, d1 = o1 - mean2;
  float var2 = blk_sum(d0 * d0 + d1 * d1, red) * (1.f / 256.f);
  float is = rsqrtf(var2 + 1e-5f);
  O[(size_t)row * 256 + tid]       = d0 * is * g2[tid] + bb2[tid];
  O[(size_t)row * 256 + tid + 128] = d1 * is * g2[tid + 128] + bb2[tid + 128];
}

// out[m][d] = in[m][d] * conf[m]   (pre-scale for PMA V projection)
__global__ void rowscale_kernel(const float* __restrict__ X, const float* __restrict__ conf,
                                float* __restrict__ O) {
  int row = blockIdx.x, tid = threadIdx.x;
  O[(size_t)row * 256 + tid] = X[(size_t)row * 256 + tid] * conf[row];
}

// sm[b*2+s][d] = seeds[s][d]
__global__ void bcast_seeds_kernel(const float* __restrict__ seeds, float* __restrict__ sm) {
  int row = blockIdx.x, tid = threadIdx.x;      // 32 rows x 256
  sm[(size_t)row * 256 + tid] = seeds[(row & 1) * 256 + tid];
}

// pooled[b][d] = 0.5*(y[2b][d] + y[2b+1][d])
__global__ void seedmean_kernel(const float* __restrict__ Y, float* __restrict__ P) {
  int b = blockIdx.x, tid = threadIdx.x;
  P[(size_t)b * 256 + tid] =
      0.5f * (Y[(size_t)(2 * b) * 256 + tid] + Y[(size_t)(2 * b + 1) * 256 + tid]);
}

// value MLP: z=[pr|pt|tier0] (546) -> gelu(LN(l1)) -> gelu(LN(l2)) -> l3
__global__ void vmlp_kernel(const float* __restrict__ pr, const float* __restrict__ pt,
                            const float* __restrict__ t0,
                            const float* __restrict__ w1, const float* __restrict__ b1,
                            const float* __restrict__ g1, const float* __restrict__ bb1,
                            const float* __restrict__ w2, const float* __restrict__ b2,
                            const float* __restrict__ g2, const float* __restrict__ bb2,
                            const float* __restrict__ w3, const float* __restrict__ b3,
                            float* __restrict__ out) {
  __shared__ float z[546];
  __shared__ float hh[256];
  __shared__ float red[256];
  int b = blockIdx.x, tid = threadIdx.x;        // 256 threads
  z[tid]       = pr[(size_t)b * 256 + tid];
  z[256 + tid] = pt[(size_t)b * 256 + tid];
  if (tid < 34) z[512 + tid] = t0[b * 34 + tid];
  __syncthreads();

  float s = b1[tid];
  for (int i = 0; i < 546; ++i) s += w1[(size_t)tid * 546 + i] * z[i];
  float mean = blk_sum(s, red) * (1.f / 256.f);
  float dd = s - mean;
  float var = blk_sum(dd * dd, red) * (1.f / 256.f);
  s = gelu_f(dd * rsqrtf(var + 1e-5f) * g1[tid] + bb1[tid]);
  hh[tid] = s;
  __syncthreads();

  float s2 = b2[tid];
  for (int i = 0; i < 256; ++i) s2 += w2[(size_t)tid * 256 + i] * hh[i];
  float mean2 = blk_sum(s2, red) * (1.f / 256.f);
  float d2 = s2 - mean2;
  float var2 = blk_sum(d2 * d2, red) * (1.f / 256.f);
  s2 = gelu_f(d2 * rsqrtf(var2 + 1e-5f) * g2[tid] + bb2[tid]);

  float tot = blk_sum(s2 * w3[tid], red);
  if (tid == 0) out[b] = tot + b3[0];
}

// ---------------------------------------------------------------------------
// Host side
// ---------------------------------------------------------------------------
struct MhaP { const float *wqw,*wqb,*wkw,*wkb,*wvw,*wvb,*wow,*wob; };

// leaves at base i: wk.b, wk.w, wo.b, wo.w, wq.b, wq.w, wv.b, wv.w
static MhaP mha_at(void* const* din, int i) {
  MhaP m;
  m.wkb = (const float*)din[i + 0]; m.wkw = (const float*)din[i + 1];
  m.wob = (const float*)din[i + 2]; m.wow = (const float*)din[i + 3];
  m.wqb = (const float*)din[i + 4]; m.wqw = (const float*)din[i + 5];
  m.wvb = (const float*)din[i + 6]; m.wvw = (const float*)din[i + 7];
  return m;
}

static void run_gemm(hipStream_t s, const float* A, const float* W, const float* bias,
                     float* C, int M, int N, int K, int act) {
  dim3 g(N / 64, M / 32), blk(32);
  gemm_wmma_kernel<<<g, blk, 0, s>>>(A, W, bias, C, M, N, K, act);
}

// SAB layer leaves at base L: f1.b,f1.w,f2.b,f2.w,ln1.b,ln1.g,ln2.b,ln2.g,mha(8)
static void run_sab(hipStream_t s, float* x, int Nq, const int* mask, const float* conf,
                    void* const* din, int L, float* qb, float* kb, float* vb,
                    float* ao, float* ffn) {
  const int Bv = 16, M = Bv * Nq;
  const float* f1b = (const float*)din[L + 0]; const float* f1w = (const float*)din[L + 1];
  const float* f2b = (const float*)din[L + 2]; const float* f2w = (const float*)din[L + 3];
  const float* l1b = (const float*)din[L + 4]; const float* l1g = (const float*)din[L + 5];
  const float* l2b = (const float*)din[L + 6]; const float* l2g = (const float*)din[L + 7];
  MhaP mh = mha_at(din, L + 8);
  run_gemm(s, x, mh.wqw, mh.wqb, qb, M, 256, 256, 0);
  run_gemm(s, x, mh.wkw, mh.wkb, kb, M, 256, 256, 0);
  run_gemm(s, x, mh.wvw, mh.wvb, vb, M, 256, 256, 0);
  attn_kernel<<<dim3(Bv * 8 * Nq), dim3(256), 0, s>>>(qb, kb, vb, conf, mask, ao, Nq, Nq, 1);
  run_gemm(s, ao, mh.wow, mh.wob, qb, M, 256, 256, 0);
  ln_kernel<<<M, 256, 0, s>>>(x, qb, l1g, l1b, x);
  run_gemm(s, x, f1w, f1b, ffn, M, 512, 256, 1);
  run_gemm(s, ffn, f2w, f2b, qb, M, 256, 512, 0);
  ln_kernel<<<M, 256, 0, s>>>(x, qb, l2g, l2b, x);
}

// PMA leaves at base L: ln.b, ln.g, mha(8), seeds
static void run_pma(hipStream_t s, const float* x, int Nk, const int* mask,
                    const float* conf, void* const* din, int L,
                    float* kb, float* vb, float* xscaled,
                    float* sm, float* pq, float* pao, float* pwo, float* pln,
                    float* pooled) {
  const int Bv = 16, M = Bv * Nk;
  const float* lnb = (const float*)din[L + 0]; const float* lng = (const float*)din[L + 1];
  MhaP mh = mha_at(din, L + 2);
  const float* seeds = (const float*)din[L + 10];
  bcast_seeds_kernel<<<32, 256, 0, s>>>(seeds, sm);
  run_gemm(s, sm, mh.wqw, mh.wqb, pq, 32, 256, 256, 0);
  run_gemm(s, x, mh.wkw, mh.wkb, kb, M, 256, 256, 0);
  rowscale_kernel<<<M, 256, 0, s>>>(x, conf, xscaled);   // vals = x*conf BEFORE wv
  run_gemm(s, xscaled, mh.wvw, mh.wvb, vb, M, 256, 256, 0);
  attn_kernel<<<dim3(Bv * 8 * 2), dim3(256), 0, s>>>(pq, kb, vb, conf, mask, pao, 2, Nk, 0);
  run_gemm(s, pao, mh.wow, mh.wob, pwo, 32, 256, 256, 0);
  ln_kernel<<<32, 256, 0, s>>>(pwo, nullptr, lng, lnb, pln);
  seedmean_kernel<<<16, 256, 0, s>>>(pln, pooled);
}

extern "C" void kernel_launch(void* const* d_in, const int* in_sizes, int n_in,
                              void* d_out, int out_size, void* d_ws, size_t ws_size,
                              hipStream_t stream) {
  const float* tier0 = (const float*)d_in[0];
  const float* rfeat = (const float*)d_in[1];
  const float* tfeat = (const float*)d_in[2];
  const float* rconf = (const float*)d_in[3];
  const float* tconf = (const float*)d_in[4];
  const int*   rmask = (const int*)d_in[5];
  const int*   tmask = (const int*)d_in[6];

  // params leaves flattened jax-style (sorted dict keys), base P = 7:
  //  cross: r2t mha @+0..7, rn.b +8, rn.g +9, t2r mha @+10..17, tn.b +18, tn.g +19
  //  robot_pma @+20 (ln.b, ln.g, mha x8, seeds)
  //  robot_proj @+31 (l1.b,l1.w,l2.b,l2.w,ln1.b,ln1.g,ln2.b,ln2.g)
  //  robot_sab[0] @+39, robot_sab[1] @+55
  //  track_pma @+71, track_proj @+82, track_sab[0] @+90, track_sab[1] @+106
  //  vmlp @+122 (l1.b,l1.w,l2.b,l2.w,l3.b,l3.w,ln1.b,ln1.g,ln2.b,ln2.g)
  const int P = 7;
  #define PL(i) ((const float*)d_in[P + (i)])

  float* ws = (float*)d_ws;
  size_t off = 0;
  float* r   = ws + off; off += (size_t)4096 * 256;
  float* t   = ws + off; off += (size_t)16384 * 256;
  float* qb  = ws + off; off += (size_t)16384 * 256;
  float* kb  = ws + off; off += (size_t)16384 * 256;
  float* vb  = ws + off; off += (size_t)16384 * 256;
  float* ao  = ws + off; off += (size_t)16384 * 256;
  float* ffn = ws + off; off += (size_t)16384 * 512;
  float* sm  = ws + off; off += 32 * 256;
  float* pq  = ws + off; off += 32 * 256;
  float* pao = ws + off; off += 32 * 256;
  float* pwo = ws + off; off += 32 * 256;
  float* pln = ws + off; off += 32 * 256;
  float* prp = ws + off; off += 16 * 256;
  float* ptp = ws + off; off += 16 * 256;
  float* cr  = ffn;                       // cross outputs live in the FFN buffer
  float* ct  = ffn + (size_t)4096 * 256;

  // 1. input projections
  proj_kernel<<<4096, 128, 0, stream>>>(rfeat, 6,
      PL(32), PL(31), PL(36), PL(35), PL(34), PL(33), PL(38), PL(37), r);
  proj_kernel<<<16384, 128, 0, stream>>>(tfeat, 7,
      PL(83), PL(82), PL(87), PL(86), PL(85), PL(84), PL(89), PL(88), t);

  // 2. SAB stacks
  run_sab(stream, r, 256,  rmask, rconf, d_in, P + 39,  qb, kb, vb, ao, ffn);
  run_sab(stream, r, 256,  rmask, rconf, d_in, P + 55,  qb, kb, vb, ao, ffn);
  run_sab(stream, t, 1024, tmask, tconf, d_in, P + 90,  qb, kb, vb, ao, ffn);
  run_sab(stream, t, 1024, tmask, tconf, d_in, P + 106, qb, kb, vb, ao, ffn);

  // 3. cross attention (both use pre-cross r and t)
  {
    MhaP r2t = mha_at(d_in, P + 0);
    MhaP t2r = mha_at(d_in, P + 10);
    run_gemm(stream, r, r2t.wqw, r2t.wqb, qb, 4096, 256, 256, 0);
    run_gemm(stream, t, r2t.wkw, r2t.wkb, kb, 16384, 256, 256, 0);
    run_gemm(stream, t, r2t.wvw, r2t.wvb, vb, 16384, 256, 256, 0);
    attn_kernel<<<dim3(16 * 8 * 256), dim3(256), 0, stream>>>(qb, kb, vb, tconf, tmask, ao, 256, 1024, 2);
    run_gemm(stream, ao, r2t.wow, r2t.wob, cr, 4096, 256, 256, 0);

    run_gemm(stream, t, t2r.wqw, t2r.wqb, qb, 16384, 256, 256, 0);
    run_gemm(stream, r, t2r.wkw, t2r.wkb, kb, 4096, 256, 256, 0);
    run_gemm(stream, r, t2r.wvw, t2r.wvb, vb, 4096, 256, 256, 0);
    attn_kernel<<<dim3(16 * 8 * 1024), dim3(256), 0, stream>>>(qb, kb, vb, rconf, rmask, ao, 1024, 256, 2);
    run_gemm(stream, ao, t2r.wow, t2r.wob, ct, 16384, 256, 256, 0);

    ln_kernel<<<4096,  256, 0, stream>>>(r, cr, PL(9),  PL(8),  r);   // rn
    ln_kernel<<<16384, 256, 0, stream>>>(t, ct, PL(19), PL(18), t);   // tn
  }

  // 4. PMA pooling (xscaled reuses ao)
  run_pma(stream, r, 256,  rmask, rconf, d_in, P + 20, kb, vb, ao, sm, pq, pao, pwo, pln, prp);
  run_pma(stream, t, 1024, tmask, tconf, d_in, P + 71, kb, vb, ao, sm, pq, pao, pwo, pln, ptp);

  // 5. value MLP -> d_out[16]
  vmlp_kernel<<<16, 256, 0, stream>>>(prp, ptp, tier0,
      PL(123), PL(122), PL(129), PL(128),   // l1.w, l1.b, ln1.g, ln1.b
      PL(125), PL(124), PL(131), PL(130),   // l2.w, l2.b, ln2.g, ln2.b
      PL(127), PL(126),                     // l3.w, l3.b
      (float*)d_out);
  #undef PL
}